// QuadMamba_18829136626306
// MI455X (gfx1250) — compile-verified
//
#include <hip/hip_runtime.h>
#include <cstdint>
#include <cstddef>

// Problem constants (from reference)
#define BB 4
#define HH 32
#define WWD 256
#define CC 96
#define DI 192
#define KK 2
#define NN 16
#define RR 6
#define LL (HH*WWD)          // 8192
#define BL (BB*LL)           // 32768
#define NOUT (2*DI)          // 384
#define CDBL (RR + 2*NN)     // 38
#define NC 128               // scan chunks
#define CL (LL/NC)           // 64 steps per chunk

typedef float v2f __attribute__((ext_vector_type(2)));
typedef float v8f __attribute__((ext_vector_type(8)));

// f32 WMMA: D(16x16) = A(16x4) x B(4x16) + C.
// (neg_a, A, neg_b, B, c_mod, C, reuse_a, reuse_b)
__device__ __forceinline__ v8f wmma4(v2f a, v2f b, v8f c) {
  return __builtin_amdgcn_wmma_f32_16x16x4_f32(false, a, false, b, (short)0, c,
                                               false, false);
}

__device__ __forceinline__ float silu_f(float x) {
  return x / (1.0f + __expf(-x));
}
__device__ __forceinline__ float softplus_f(float x) {
  return (x > 20.0f) ? x : log1pf(__expf(x));
}

// ---------------------------------------------------------------------------
// Kernel 1: xz = x @ W_in^T   (M=BL, N=384, K=96), f32 WMMA 16x16x4
// ---------------------------------------------------------------------------
__global__ void __launch_bounds__(256) k_inproj(const float* __restrict__ x,
                                                const float* __restrict__ W_in,
                                                float* __restrict__ xz) {
  __shared__ float sX[16][CC + 1];
  const int rowBase = blockIdx.x * 16;
  const int tid = threadIdx.x;
  for (int i = tid; i < 16 * CC; i += 256) {
    int m = i / CC, kk = i % CC;
    sX[m][kk] = x[(size_t)(rowBase + m) * CC + kk];
  }
  __syncthreads();
  const int wave = tid >> 5, lane = tid & 31;
  const int lid = lane & 15, half = lane >> 4;
  for (int nt = wave; nt < NOUT / 16; nt += 8) {
    const int nbase = nt * 16;
    const float* wrow = W_in + (size_t)(nbase + lid) * CC;
    v8f acc = {};
    for (int k0 = 0; k0 < CC; k0 += 4) {
      v2f a; a.x = sX[lid][k0 + 2 * half]; a.y = sX[lid][k0 + 2 * half + 1];
      v2f b; b.x = wrow[k0 + 2 * half];    b.y = wrow[k0 + 2 * half + 1];
      acc = wmma4(a, b, acc);
    }
    float* orow = xz + (size_t)(rowBase + half * 8) * NOUT + nbase + lid;
#pragma unroll
    for (int v = 0; v < 8; ++v) orow[(size_t)v * NOUT] = acc[v];
  }
}

// ---------------------------------------------------------------------------
// Kernel 2: depthwise 3x3 conv (SAME) + bias + SiLU, channels-last.
// ---------------------------------------------------------------------------
__global__ void __launch_bounds__(256) k_dwconv(const float* __restrict__ xz,
                                                const float* __restrict__ conv_w,
                                                const float* __restrict__ conv_b,
                                                float* __restrict__ xconv) {
  int idx = blockIdx.x * 256 + threadIdx.x;
  if (idx >= BB * HH * WWD * DI) return;
  int c = idx % DI; int t = idx / DI;
  int w = t % WWD; t /= WWD;
  int h = t % HH;  int b = t / HH;
  float s = conv_b[c];
#pragma unroll
  for (int dy = -1; dy <= 1; ++dy) {
    int hh = h + dy; if (hh < 0 || hh >= HH) continue;
#pragma unroll
    for (int dx = -1; dx <= 1; ++dx) {
      int ww = w + dx; if (ww < 0 || ww >= WWD) continue;
      s += xz[(size_t)((b * HH + hh) * WWD + ww) * NOUT + c] *
           conv_w[c * 9 + (dy + 1) * 3 + (dx + 1)];
    }
  }
  xconv[(size_t)((b * HH + h) * WWD + w) * DI + c] = silu_f(s);
}

// ---------------------------------------------------------------------------
// Kernel 3: x_dbl = einsum('bkdl,kcd->bkcl')  (N=38 padded to 48, K=192)
// ---------------------------------------------------------------------------
__global__ void __launch_bounds__(256) k_xdbl(const float* __restrict__ xconv,
                                              const float* __restrict__ xpw,
                                              float* __restrict__ x_dbl) {
  const int tid = threadIdx.x, wave = tid >> 5, lane = tid & 31;
  const int lid = lane & 15, half = lane >> 4;
  const int tile = blockIdx.x * 8 + wave;     // 0 .. B*K*L/16 - 1
  const int lt = tile % (LL / 16);
  const int bk = tile / (LL / 16);
  const int k = bk % KK, b = bk / KK;
  const int lbase = lt * 16;
  const int l = lbase + lid;
  const int lidx = k ? (LL - 1 - l) : l;      // scan order -> spatial
  const float* arow = xconv + (size_t)(b * LL + lidx) * DI;
  v8f acc[3] = {};
  for (int k0 = 0; k0 < DI; k0 += 4) {
    v2f a; a.x = arow[k0 + 2 * half]; a.y = arow[k0 + 2 * half + 1];
#pragma unroll
    for (int t = 0; t < 3; ++t) {
      int c2 = t * 16 + lid;
      v2f bb;
      if (c2 < CDBL) {
        const float* wrow = xpw + (size_t)(k * CDBL + c2) * DI;
        bb.x = wrow[k0 + 2 * half]; bb.y = wrow[k0 + 2 * half + 1];
      } else { bb.x = 0.f; bb.y = 0.f; }
      acc[t] = wmma4(a, bb, acc[t]);
    }
  }
#pragma unroll
  for (int t = 0; t < 3; ++t) {
    int col = t * 16 + lid;
    if (col < CDBL) {
      float* orow = x_dbl + ((size_t)bk * LL + lbase + half * 8) * CDBL + col;
#pragma unroll
      for (int v = 0; v < 8; ++v) orow[(size_t)v * CDBL] = acc[t][v];
    }
  }
}

// ---------------------------------------------------------------------------
// Kernel 4: dts = dts_r @ dt_projs_weight^T (K=6 padded to 8 -> 2 WMMA steps)
// then delta = softplus(dts + bias), layout (B,K,L,DI)
// ---------------------------------------------------------------------------
__global__ void __launch_bounds__(256) k_dtproj(const float* __restrict__ x_dbl,
                                                const float* __restrict__ dtw,
                                                const float* __restrict__ dtb,
                                                float* __restrict__ delta) {
  const int tid = threadIdx.x, wave = tid >> 5, lane = tid & 31;
  const int lid = lane & 15, half = lane >> 4;
  const int tile = blockIdx.x;                 // one l-tile per block
  const int lt = tile % (LL / 16);
  const int bk = tile / (LL / 16);
  const int k = bk % KK;
  const int lbase = lt * 16;
  const float* arow = x_dbl + ((size_t)bk * LL + lbase + lid) * CDBL;
  v2f a0, a1;
  a0.x = arow[2 * half]; a0.y = arow[2 * half + 1];       // r = 0..3
  a1.x = (half == 0) ? arow[4] : 0.f;                     // r = 4,5; pad 6,7
  a1.y = (half == 0) ? arow[5] : 0.f;
  for (int nt = wave; nt < DI / 16; nt += 8) {
    const int d = nt * 16 + lid;
    const float* wrow = dtw + (size_t)(k * DI + d) * RR;
    v2f b0, b1;
    b0.x = wrow[2 * half]; b0.y = wrow[2 * half + 1];
    b1.x = (half == 0) ? wrow[4] : 0.f;
    b1.y = (half == 0) ? wrow[5] : 0.f;
    v8f acc = {};
    acc = wmma4(a0, b0, acc);
    acc = wmma4(a1, b1, acc);
    const float bias = dtb[k * DI + d];
    float* orow = delta + ((size_t)bk * LL + lbase + half * 8) * DI + d;
#pragma unroll
    for (int v = 0; v < 8; ++v) orow[(size_t)v * DI] = softplus_f(acc[v] + bias);
  }
}

// ---------------------------------------------------------------------------
// Scan, 3-pass chunked formulation. Recurrence per (b,k,d,n):
//   h_l = a_l * h_{l-1} + b_l,  a_l = exp(dlt_l*A_n),  b_l = dlt_l*u_l*Bs_l[n]
// Pass 1: per chunk c, from h=0 compute (aP = prod a, hloc = local scan tail).
// Pass 2: per chain, sequentially compose chunks; overwrite hloc with h0(c).
// Pass 3: per chunk, rerun local scan from h0 and emit y (+ u*D).
// Chunk summaries layout: [((bk*NC + c)*NN + n)*DI + d]  (coalesced in d).
// ---------------------------------------------------------------------------
__global__ void __launch_bounds__(256) k_scan_pass1(
    const float* __restrict__ xconv, const float* __restrict__ delta,
    const float* __restrict__ x_dbl, const float* __restrict__ A_logs,
    float* __restrict__ chunkA, float* __restrict__ chunkH) {
  const int gid = blockIdx.x * 256 + threadIdx.x;   // exact grid
  const int d = gid % DI;
  const int c = (gid / DI) % NC;
  const int bk = gid / (DI * NC);
  const int k = bk % KK, b = bk / KK;
  float Av[NN];
#pragma unroll
  for (int n = 0; n < NN; ++n)
    Av[n] = -__expf(A_logs[(size_t)(k * DI + d) * NN + n]);
  float aP[NN], h[NN];
#pragma unroll
  for (int n = 0; n < NN; ++n) { aP[n] = 1.f; h[n] = 0.f; }
  const float* xc = xconv + (size_t)b * LL * DI;
  const float* dl = delta + (size_t)bk * LL * DI;
  const float* xd = x_dbl + (size_t)bk * LL * CDBL;
  const int l0 = c * CL;
  for (int l = l0; l < l0 + CL; ++l) {
    const int lidx = k ? (LL - 1 - l) : l;
    const float u = xc[(size_t)lidx * DI + d];
    const float dlt = dl[(size_t)l * DI + d];
    const float* row = xd + (size_t)l * CDBL;
    const float du = dlt * u;
#pragma unroll
    for (int n = 0; n < NN; ++n) {
      const float dA = __expf(dlt * Av[n]);
      aP[n] *= dA;
      h[n] = dA * h[n] + du * row[RR + n];
    }
  }
#pragma unroll
  for (int n = 0; n < NN; ++n) {
    const size_t idx = ((size_t)(bk * NC + c) * NN + n) * DI + d;
    chunkA[idx] = aP[n];
    chunkH[idx] = h[n];
  }
}

__global__ void __launch_bounds__(256) k_scan_pass2(
    const float* __restrict__ chunkA, float* __restrict__ chunkH) {
  const int gid = blockIdx.x * 256 + threadIdx.x;   // 1536 threads exact
  const int d = gid % DI;
  const int bk = gid / DI;
  float hrun[NN];
#pragma unroll
  for (int n = 0; n < NN; ++n) hrun[n] = 0.f;
  for (int c = 0; c < NC; ++c) {
#pragma unroll
    for (int n = 0; n < NN; ++n) {
      const size_t idx = ((size_t)(bk * NC + c) * NN + n) * DI + d;
      const float a = chunkA[idx];
      const float hl = chunkH[idx];
      chunkH[idx] = hrun[n];               // h0 for this chunk
      hrun[n] = a * hrun[n] + hl;
    }
  }
}

__global__ void __launch_bounds__(256) k_scan_pass3(
    const float* __restrict__ xconv, const float* __restrict__ delta,
    const float* __restrict__ x_dbl, const float* __restrict__ A_logs,
    const float* __restrict__ Ds, const float* __restrict__ chunkH,
    float* __restrict__ ys) {
  const int gid = blockIdx.x * 256 + threadIdx.x;   // exact grid
  const int d = gid % DI;
  const int c = (gid / DI) % NC;
  const int bk = gid / (DI * NC);
  const int k = bk % KK, b = bk / KK;
  float Av[NN];
#pragma unroll
  for (int n = 0; n < NN; ++n)
    Av[n] = -__expf(A_logs[(size_t)(k * DI + d) * NN + n]);
  const float Dv = Ds[k * DI + d];
  float h[NN];
#pragma unroll
  for (int n = 0; n < NN; ++n)
    h[n] = chunkH[((size_t)(bk * NC + c) * NN + n) * DI + d];
  const float* xc = xconv + (size_t)b * LL * DI;
  const float* dl = delta + (size_t)bk * LL * DI;
  const float* xd = x_dbl + (size_t)bk * LL * CDBL;
  float* yo = ys + (size_t)bk * LL * DI;
  const int l0 = c * CL;
  for (int l = l0; l < l0 + CL; ++l) {
    const int lidx = k ? (LL - 1 - l) : l;
    const float u = xc[(size_t)lidx * DI + d];
    const float dlt = dl[(size_t)l * DI + d];
    const float* row = xd + (size_t)l * CDBL;
    const float du = dlt * u;
    float y = 0.f;
#pragma unroll
    for (int n = 0; n < NN; ++n) {
      const float dA = __expf(dlt * Av[n]);
      h[n] = dA * h[n] + du * row[RR + n];         // Bs
      y += h[n] * row[RR + NN + n];                // Cs
    }
    yo[(size_t)l * DI + d] = y + u * Dv;
  }
}

// ---------------------------------------------------------------------------
// Kernel 6: out = ((ys0 + flip(ys1)) * silu(z)) @ W_out^T  (N=96, K=192)
// ---------------------------------------------------------------------------
__global__ void __launch_bounds__(256) k_outproj(const float* __restrict__ ys,
                                                 const float* __restrict__ xz,
                                                 const float* __restrict__ W_out,
                                                 float* __restrict__ out) {
  __shared__ float sY[16][DI + 1];
  const int rowBase = blockIdx.x * 16;
  const int b = rowBase / LL;
  const int lb = rowBase % LL;
  const int tid = threadIdx.x;
  for (int i = tid; i < 16 * DI; i += 256) {
    int m = i / DI, dd = i % DI;
    int l = lb + m;
    float y0 = ys[((size_t)(b * KK + 0) * LL + l) * DI + dd];
    float y1 = ys[((size_t)(b * KK + 1) * LL + (LL - 1 - l)) * DI + dd];
    float zv = xz[(size_t)(b * LL + l) * NOUT + DI + dd];
    sY[m][dd] = (y0 + y1) * silu_f(zv);
  }
  __syncthreads();
  const int wave = tid >> 5, lane = tid & 31;
  const int lid = lane & 15, half = lane >> 4;
  for (int nt = wave; nt < CC / 16; nt += 8) {   // 6 N-tiles
    const int nbase = nt * 16;
    const float* wrow = W_out + (size_t)(nbase + lid) * DI;
    v8f acc = {};
    for (int k0 = 0; k0 < DI; k0 += 4) {
      v2f a;  a.x = sY[lid][k0 + 2 * half];  a.y = sY[lid][k0 + 2 * half + 1];
      v2f bb; bb.x = wrow[k0 + 2 * half];    bb.y = wrow[k0 + 2 * half + 1];
      acc = wmma4(a, bb, acc);
    }
    float* orow = out + (size_t)(rowBase + half * 8) * CC + nbase + lid;
#pragma unroll
    for (int v = 0; v < 8; ++v) orow[(size_t)v * CC] = acc[v];
  }
}

// ---------------------------------------------------------------------------
extern "C" void kernel_launch(void* const* d_in, const int* in_sizes, int n_in,
                              void* d_out, int out_size, void* d_ws, size_t ws_size,
                              hipStream_t stream) {
  (void)in_sizes; (void)n_in; (void)out_size; (void)ws_size;
  const float* x      = (const float*)d_in[0];
  const float* W_in   = (const float*)d_in[1];
  const float* conv_w = (const float*)d_in[2];
  const float* conv_b = (const float*)d_in[3];
  const float* xpw    = (const float*)d_in[4];
  const float* dtw    = (const float*)d_in[5];
  const float* dtb    = (const float*)d_in[6];
  const float* A_logs = (const float*)d_in[7];
  const float* Ds     = (const float*)d_in[8];
  const float* W_out  = (const float*)d_in[9];
  float* out = (float*)d_out;

  char* ws = (char*)d_ws;
  size_t off = 0;
  auto alloc = [&](size_t nbytes) {
    char* p = ws + off;
    off = (off + nbytes + 255) & ~(size_t)255;
    return (float*)p;
  };
  float* xz     = alloc((size_t)BL * NOUT * sizeof(float));          // 48 MB
  float* xconv  = alloc((size_t)BL * DI * sizeof(float));            // 24 MB
  float* x_dbl  = alloc((size_t)BB * KK * LL * CDBL * sizeof(float));//  9.5 MB
  float* delta  = alloc((size_t)BB * KK * LL * DI * sizeof(float));  // 48 MB
  float* ysb    = alloc((size_t)BB * KK * LL * DI * sizeof(float));  // 48 MB
  float* chunkA = alloc((size_t)BB * KK * DI * NC * NN * sizeof(float)); // 12.6 MB
  float* chunkH = alloc((size_t)BB * KK * DI * NC * NN * sizeof(float)); // 12.6 MB

  k_inproj<<<BL / 16, 256, 0, stream>>>(x, W_in, xz);
  k_dwconv<<<(BB * HH * WWD * DI + 255) / 256, 256, 0, stream>>>(xz, conv_w,
                                                                 conv_b, xconv);
  k_xdbl<<<(BB * KK * LL / 16) / 8, 256, 0, stream>>>(xconv, xpw, x_dbl);
  k_dtproj<<<BB * KK * LL / 16, 256, 0, stream>>>(x_dbl, dtw, dtb, delta);
  k_scan_pass1<<<(BB * KK * DI * NC) / 256, 256, 0, stream>>>(
      xconv, delta, x_dbl, A_logs, chunkA, chunkH);
  k_scan_pass2<<<(BB * KK * DI) / 256, 256, 0, stream>>>(chunkA, chunkH);
  k_scan_pass3<<<(BB * KK * DI * NC) / 256, 256, 0, stream>>>(
      xconv, delta, x_dbl, A_logs, Ds, chunkH, ysb);
  k_outproj<<<BL / 16, 256, 0, stream>>>(ysb, xz, W_out, out);
}